// NonMaximaSuppression3d_22840636080730
// MI455X (gfx1250) — compile-verified
//
#include <hip/hip_runtime.h>
#include <stdint.h>

// Problem geometry (fixed by the reference): x is (B=4, C=2, D=64, H=256, W=256) fp32.
#define DIM_W 256
#define DIM_H 256
#define DIM_D 64
#define CH    16             // output rows per block
#define RROWS (CH + 2)       // slab rows (halo of 1 on each side)
#define NCHUNK (DIM_H / CH)  // 16 row-chunks per plane

typedef __attribute__((ext_vector_type(4))) unsigned int u32x4;
typedef __attribute__((ext_vector_type(8))) unsigned int u32x8;

__global__ __launch_bounds__(256) void nms3d_tdm_kernel(const float* __restrict__ x,
                                                        float* __restrict__ out) {
  const int w     = threadIdx.x;                    // 0..255 == full W row
  const int chunk = blockIdx.x % NCHUNK;
  const int d     = (blockIdx.x / NCHUNK) % DIM_D;
  const int bc    = blockIdx.x / (NCHUNK * DIM_D);  // fused batch*channel
  const int h0    = chunk * CH;

  const long long planeStride = (long long)DIM_H * DIM_W;           // 65536 elements
  const long long chanBase    = (long long)bc * DIM_D * planeStride;
  float* outPlane = out + chanBase + (long long)d * planeStride;

  // Boundary planes: edge-padding duplicates the center into the neighbor set,
  // so the strict '>' mask is always false -> whole plane is zero.
  if (d == 0 || d == DIM_D - 1) {
    for (int h = h0; h < h0 + CH; ++h)
      __builtin_nontemporal_store(0.0f, &outPlane[(long long)h * DIM_W + w]);
    return;
  }

  // Slab covers input rows [r0, r0+RROWS) of planes d-1, d, d+1.
  int r0 = h0 - 1;
  if (r0 < 0) r0 = 0;
  if (r0 > DIM_H - RROWS) r0 = DIM_H - RROWS;

  __shared__ float slab[3 * RROWS * DIM_W];  // 55296 bytes

  // ---- Tensor Data Mover: one 3-D tile load (x=256, y=18 rows, z=3 planes) ----
  // Issued by wave 0 only (TDM ignores EXEC, so gate with a scalar branch).
  if (__builtin_amdgcn_readfirstlane((int)threadIdx.x) == 0) {
    const uint64_t gaddr =
        (uint64_t)(uintptr_t)(x + chanBase + (long long)(d - 1) * planeStride +
                              (long long)r0 * DIM_W);
    const uint32_t lds = (uint32_t)(uintptr_t)&slab[0];  // low 32 bits = LDS offset

    u32x4 g0;
    g0[0] = 1u;                                    // count=1 (valid), user mode
    g0[1] = lds;                                   // lds_addr (bytes)
    g0[2] = (uint32_t)gaddr;                       // global_addr[31:0]
    g0[3] = (uint32_t)(gaddr >> 32) | (2u << 30);  // global_addr[56:32] | type=2

    u32x8 g1;
    g1[0] = 2u << 16;                        // workgroup_mask=0, data_size=2 (4B)
    g1[1] = ((uint32_t)DIM_W) << 16;         // tensor_dim0[15:0] = 256
    g1[2] = ((uint32_t)RROWS) << 16;         // tensor_dim1[15:0] = 18
    g1[3] = ((uint32_t)DIM_W) << 16;         // tile_dim0 = 256
    g1[4] = (uint32_t)RROWS | (3u << 16);    // tile_dim1=18 | tile_dim2=3
    g1[5] = (uint32_t)DIM_W;                 // tensor_dim0_stride[31:0] = 256 (row pitch)
    g1[6] = 0u;                              // dim0_stride[47:32]=0 | dim1_stride[15:0]=0
    g1[7] = (uint32_t)(planeStride >> 16);   // dim1_stride[47:16]: 65536>>16 = 1 (plane pitch)

    u32x4 g2;
    g2[0] = 3u;  // tensor_dim2 = 3 planes
    g2[1] = 0u;  // tensor_dim3 (unused, iterate off)
    g2[2] = 0u;  // tensor_dim2_stride[31:0] (unused: tile_dim3=0)
    g2[3] = 0u;  // tensor_dim2_stride[47:32] | tile_dim3=0
    u32x4 g3 = {0u, 0u, 0u, 0u};

    asm volatile("tensor_load_to_lds %0, %1, %2, %3"
                 :
                 : "s"(g0), "s"(g1), "s"(g2), "s"(g3)
                 : "memory");
    __builtin_amdgcn_s_wait_tensorcnt(0);
  }
  __syncthreads();

  const float* s0 = &slab[0];                  // plane d-1
  const float* s1 = &slab[RROWS * DIM_W];      // plane d
  const float* s2 = &slab[2 * RROWS * DIM_W];  // plane d+1

  const int wl = (w == 0) ? 0 : w - 1;             // clamped; w=0/255 lanes output 0 anyway
  const int wr = (w == DIM_W - 1) ? DIM_W - 1 : w + 1;

  // Horizontal 3-max of a slab row.
  auto rowA = [&](const float* sp, int lr) -> float {
    float a = sp[lr * DIM_W + wl];
    float b = sp[lr * DIM_W + w];
    float c = sp[lr * DIM_W + wr];
    return fmaxf(a, fmaxf(b, c));
  };
  // Same, plus center-excluded horizontal max and the raw center value.
  auto rowABX = [&](const float* sp, int lr, float& A, float& B, float& X) {
    float a = sp[lr * DIM_W + wl];
    float b = sp[lr * DIM_W + w];
    float c = sp[lr * DIM_W + wr];
    A = fmaxf(a, fmaxf(b, c));
    B = fmaxf(a, c);
    X = b;
  };

  const int hstart = (h0 > 1) ? h0 : 1;
  const int hend   = (h0 + CH - 1 < DIM_H - 2) ? (h0 + CH - 1) : (DIM_H - 2);
  const bool wedge = (w == 0) || (w == DIM_W - 1);

  // Prime rotation with rows hstart-1 and hstart.
  float am0, am1, ac0, ac1, ap0, ap1, bcur, xcur;
  {
    int lr = hstart - 1 - r0;
    am0 = rowA(s0, lr);
    ap0 = rowA(s2, lr);
    ac0 = rowA(s1, lr);
    float dumB, dumX;
    lr = hstart - r0;
    am1 = rowA(s0, lr);
    ap1 = rowA(s2, lr);
    rowABX(s1, lr, ac1, bcur, xcur);
    (void)dumB; (void)dumX;
  }

  for (int h = hstart; h <= hend; ++h) {
    const int nr = h + 1 - r0;
    float am2 = rowA(s0, nr);
    float ap2 = rowA(s2, nr);
    float ac2, bn, xn;
    rowABX(s1, nr, ac2, bn, xn);

    // 26-neighbor max: full 3x3 of planes d-1/d+1, center-excluded 3x3 of plane d.
    float vm = fmaxf(am0, fmaxf(am1, am2));
    float vp = fmaxf(ap0, fmaxf(ap1, ap2));
    float pc = fmaxf(ac0, fmaxf(bcur, ac2));
    float mx = fmaxf(vm, fmaxf(vp, pc));

    float o = (!wedge && (xcur > mx)) ? xcur : 0.0f;
    __builtin_nontemporal_store(o, &outPlane[(long long)h * DIM_W + w]);

    am0 = am1; am1 = am2;
    ap0 = ap1; ap1 = ap2;
    ac0 = ac1; ac1 = ac2;
    bcur = bn; xcur = xn;
  }

  // Zero boundary rows owned by this chunk.
  if (h0 == 0)
    __builtin_nontemporal_store(0.0f, &outPlane[w]);
  if (h0 + CH == DIM_H)
    __builtin_nontemporal_store(0.0f, &outPlane[(long long)(DIM_H - 1) * DIM_W + w]);
}

extern "C" void kernel_launch(void* const* d_in, const int* in_sizes, int n_in,
                              void* d_out, int out_size, void* d_ws, size_t ws_size,
                              hipStream_t stream) {
  const float* x = (const float*)d_in[0];
  float* out = (float*)d_out;
  const int bcCount = in_sizes[0] / (DIM_D * DIM_H * DIM_W);  // B*C = 8
  const int blocks = bcCount * DIM_D * NCHUNK;                // 8192
  nms3d_tdm_kernel<<<blocks, 256, 0, stream>>>(x, out);
}